// AttenDecoder_47553877901522
// MI455X (gfx1250) — compile-verified
//
#include <hip/hip_runtime.h>
#include <math.h>

// Sizes (fixed by the reference)
#define B_   64
#define T_   32
#define S_   64
#define E_   128
#define H_   256
#define V_   32000
#define KGI  640   // 3E + H
#define NG   768   // 3H
#define MR   2048  // T*B rows of the vocab GEMM

typedef __attribute__((ext_vector_type(2))) float v2f;
typedef __attribute__((ext_vector_type(8))) float v8f;

// ---------------------------------------------------------------------------
// WMMA microkernel body: C[64 x 128] += A[64 x KD] * B[KD x 128]
//   A staged in LDS (row-major, row stride KD)
//   B = weight matrix W[n][k], row-major with row stride KD (B^T access)
//   8 waves per block, wave w owns columns [ng*128 + w*16, +16)
//   each wave accumulates 4 M-tiles (M=64) so each B fragment feeds 4 WMMAs
// fp32 WMMA fragment layout (ISA 7.12.2):
//   A 16x4: lane = m + 16*(k>>1), vgpr = k&1  -> lane loads b64 pair (k,k+1)
//   B 4x16: lane = n + 16*(k>>1), vgpr = k&1
//   C/D:    vgpr j -> (m = j + 8*(lane>>4), n = lane&15)
// ---------------------------------------------------------------------------
template <int KD>
__device__ __forceinline__ void wmma_panel(const float* __restrict__ smemA,
                                           const float* __restrict__ Wmat,
                                           int n, int lane,
                                           v8f& a0, v8f& a1, v8f& a2, v8f& a3) {
  const int ln   = lane & 15;
  const int koff = (lane >> 4) << 1;  // 0 or 2
  const float* Wrow = Wmat + (size_t)n * KD + koff;
  const float* A    = smemA + ln * KD + koff;
#pragma unroll 8
  for (int k0 = 0; k0 < KD; k0 += 4) {
    v2f bb = *(const v2f*)(Wrow + k0);
    v2f f0 = *(const v2f*)(A + k0);
    v2f f1 = *(const v2f*)(A + 16 * KD + k0);
    v2f f2 = *(const v2f*)(A + 32 * KD + k0);
    v2f f3 = *(const v2f*)(A + 48 * KD + k0);
    a0 = __builtin_amdgcn_wmma_f32_16x16x4_f32(false, f0, false, bb, (short)0, a0, false, false);
    a1 = __builtin_amdgcn_wmma_f32_16x16x4_f32(false, f1, false, bb, (short)0, a1, false, false);
    a2 = __builtin_amdgcn_wmma_f32_16x16x4_f32(false, f2, false, bb, (short)0, a2, false, false);
    a3 = __builtin_amdgcn_wmma_f32_16x16x4_f32(false, f3, false, bb, (short)0, a3, false, false);
  }
}

// ---------------------------------------------------------------------------
// Per-step GEMMs: gi = x @ W_ih^T (K=640), gh = h @ W_hh^T (K=256), M=64, N=768
// blocks 0..5 -> gi N-groups, blocks 6..11 -> gh N-groups
// ---------------------------------------------------------------------------
__global__ __launch_bounds__(256) void step_gemm(
    const float* __restrict__ target, const float* __restrict__ inp,
    const float* __restrict__ posf, const float* __restrict__ encs,
    const float* __restrict__ W_ih, const float* __restrict__ W_hh,
    const float* __restrict__ hs, float* __restrict__ gi, float* __restrict__ gh,
    int t) {
  extern __shared__ float smem[];  // up to 64*640 floats
  const int tid  = threadIdx.x;
  const bool is_gi = (blockIdx.x < 6);
  const int ng   = is_gi ? blockIdx.x : (blockIdx.x - 6);

  if (is_gi) {
    // assemble x[64][640] = [target(b,t) | posf(b) | encs(b) | inp(b,t)]
    for (int idx = tid; idx < 64 * KGI; idx += 256) {
      int b = idx / KGI, k = idx - b * KGI;
      float v;
      if (k < E_)               v = target[((size_t)b * T_ + t) * E_ + k];
      else if (k < 2 * E_)      v = posf[b * E_ + (k - E_)];
      else if (k < 2 * E_ + H_) v = encs[b * H_ + (k - 2 * E_)];
      else                      v = inp[((size_t)b * T_ + t) * E_ + (k - (2 * E_ + H_))];
      smem[idx] = v;
    }
  } else {
    const float4* s4 = (const float4*)hs;
    float4* d4 = (float4*)smem;
    for (int i = tid; i < 64 * H_ / 4; i += 256) d4[i] = s4[i];
  }
  __syncthreads();

  const int wave = tid >> 5, lane = tid & 31;
  const int n = ng * 128 + wave * 16 + (lane & 15);
  v8f a0 = {}, a1 = {}, a2 = {}, a3 = {};
  if (is_gi) wmma_panel<KGI>(smem, W_ih, n, lane, a0, a1, a2, a3);
  else       wmma_panel<H_>(smem, W_hh, n, lane, a0, a1, a2, a3);

  float* G = is_gi ? gi : gh;
  const int mb = (lane >> 4) << 3;
#pragma unroll
  for (int j = 0; j < 8; j++) {
    G[(size_t)(mb + j) * NG + n]      = a0[j];
    G[(size_t)(16 + mb + j) * NG + n] = a1[j];
    G[(size_t)(32 + mb + j) * NG + n] = a2[j];
    G[(size_t)(48 + mb + j) * NG + n] = a3[j];
  }
}

// ---------------------------------------------------------------------------
// Per-step: GRU gates -> hs, q = hs@Win^T+b, attention softmax, h_star, feat
// one block per batch row
// ---------------------------------------------------------------------------
__global__ __launch_bounds__(256) void step_att(
    const float* __restrict__ enc, const float* __restrict__ Win_w,
    const float* __restrict__ Win_b, const float* __restrict__ b_ih,
    const float* __restrict__ b_hh, const float* __restrict__ gi,
    const float* __restrict__ gh, float* __restrict__ hs,
    float* __restrict__ feat, float* __restrict__ out_hs, int t) {
  __shared__ float sh_h[H_];
  __shared__ float sh_q[2 * H_];
  __shared__ float sh_sc[S_];
  const int b = blockIdx.x, j = threadIdx.x;

  // GRU gate math (thread j owns hidden unit j)
  {
    const size_t gb = (size_t)b * NG;
    float r  = gi[gb + j] + b_ih[j] + gh[gb + j] + b_hh[j];
    float z  = gi[gb + H_ + j] + b_ih[H_ + j] + gh[gb + H_ + j] + b_hh[H_ + j];
    float in = gi[gb + 2 * H_ + j] + b_ih[2 * H_ + j];
    float hn = gh[gb + 2 * H_ + j] + b_hh[2 * H_ + j];
    r = 1.f / (1.f + expf(-r));
    z = 1.f / (1.f + expf(-z));
    float nn   = tanhf(in + r * hn);
    float hold = hs[b * H_ + j];
    float hnew = (1.f - z) * nn + z * hold;
    hs[b * H_ + j] = hnew;
    sh_h[j] = hnew;
    feat[((size_t)(t * B_ + b)) * NG + 2 * H_ + j] = hnew;
    if (t == T_ - 1) out_hs[b * H_ + j] = hnew;
  }
  __syncthreads();

  // q[512] = Win_w @ hs + Win_b  (thread j -> rows j and j+256)
  {
    float acc0 = Win_b[j], acc1 = Win_b[j + H_];
    const float* w0 = Win_w + (size_t)j * H_;
    const float* w1 = Win_w + (size_t)(j + H_) * H_;
    for (int k = 0; k < H_; k++) {
      float h = sh_h[k];
      acc0 += w0[k] * h;
      acc1 += w1[k] * h;
    }
    sh_q[j] = acc0;
    sh_q[j + H_] = acc1;
  }
  __syncthreads();

  // attn[s] = enc[b,s,:] . q
  if (j < S_) {
    const float* e = enc + ((size_t)b * S_ + j) * (2 * H_);
    float a = 0.f;
    for (int d = 0; d < 2 * H_; d++) a += e[d] * sh_q[d];
    sh_sc[j] = a;
  }
  __syncthreads();

  if (j == 0) {  // softmax over 64 entries (tiny)
    float m = -1e30f;
    for (int s = 0; s < S_; s++) m = fmaxf(m, sh_sc[s]);
    float sum = 0.f;
    for (int s = 0; s < S_; s++) { float e = expf(sh_sc[s] - m); sh_sc[s] = e; sum += e; }
    float inv = 1.f / sum;
    for (int s = 0; s < S_; s++) sh_sc[s] *= inv;
  }
  __syncthreads();

  // h_star[d] = sum_s score[s]*enc[b,s,d]; thread j -> d=j and d=j+256
  {
    float h0 = 0.f, h1 = 0.f;
    const float* e = enc + (size_t)b * S_ * (2 * H_);
    for (int s = 0; s < S_; s++) {
      float sc = sh_sc[s];
      h0 += sc * e[s * (2 * H_) + j];
      h1 += sc * e[s * (2 * H_) + H_ + j];
    }
    const size_t fb = ((size_t)(t * B_ + b)) * NG;
    feat[fb + j] = h0;
    feat[fb + H_ + j] = h1;
  }
}

// ---------------------------------------------------------------------------
// Vocab projection: y[2048 x 32000] = relu(feat[2048 x 768] @ Wout^T + b)
// grid: (32000/128 N-groups, 2048/64 M-groups); A panel (64x768=192KB) in LDS
// ---------------------------------------------------------------------------
__global__ __launch_bounds__(256) void out_gemm(
    const float* __restrict__ feat, const float* __restrict__ Wout_w,
    const float* __restrict__ Wout_b, float* __restrict__ y) {
  extern __shared__ float smem[];  // 64 x 768
  const int tid = threadIdx.x;
  const int m0 = blockIdx.y * 64;
  const int n0 = blockIdx.x * 128;

  const float4* s4 = (const float4*)(feat + (size_t)m0 * NG);
  float4* d4 = (float4*)smem;
  for (int i = tid; i < 64 * NG / 4; i += 256) d4[i] = s4[i];
  __syncthreads();

  const int wave = tid >> 5, lane = tid & 31;
  const int n = n0 + wave * 16 + (lane & 15);
  v8f a0 = {}, a1 = {}, a2 = {}, a3 = {};
  wmma_panel<NG>(smem, Wout_w, n, lane, a0, a1, a2, a3);

  const float bias = Wout_b[n];
  const int mb = m0 + ((lane >> 4) << 3);
#pragma unroll
  for (int j = 0; j < 8; j++) {
    y[(size_t)(mb + j) * V_ + n]      = fmaxf(a0[j] + bias, 0.f);
    y[(size_t)(mb + 16 + j) * V_ + n] = fmaxf(a1[j] + bias, 0.f);
    y[(size_t)(mb + 32 + j) * V_ + n] = fmaxf(a2[j] + bias, 0.f);
    y[(size_t)(mb + 48 + j) * V_ + n] = fmaxf(a3[j] + bias, 0.f);
  }
}

// ---------------------------------------------------------------------------
// Per-row online logsumexp over V=32000 (one block per row)
// ---------------------------------------------------------------------------
__global__ __launch_bounds__(256) void row_lse(const float* __restrict__ y,
                                               float* __restrict__ lse) {
  __shared__ float sm[256], ss[256];
  const int m = blockIdx.x, tid = threadIdx.x;
  const float* row = y + (size_t)m * V_;
  float mx = -1e30f, s = 0.f;
  for (int v = tid; v < V_; v += 256) {
    float x = row[v];
    if (x > mx) { s = s * expf(mx - x) + 1.f; mx = x; }
    else        { s += expf(x - mx); }
  }
  sm[tid] = mx; ss[tid] = s;
  __syncthreads();
  for (int off = 128; off > 0; off >>= 1) {
    if (tid < off) {
      float m2 = sm[tid + off], s2 = ss[tid + off];
      float M = fmaxf(sm[tid], m2);
      ss[tid] = ss[tid] * expf(sm[tid] - M) + s2 * expf(m2 - M);
      sm[tid] = M;
    }
    __syncthreads();
  }
  if (tid == 0) lse[m] = sm[0] + logf(ss[0]);
}

// in-place: y -= lse[row]  (float4; 32000 % 4 == 0 so no row crossing)
__global__ __launch_bounds__(256) void sub_lse(float* __restrict__ y,
                                               const float* __restrict__ lse) {
  const size_t i4 = (size_t)blockIdx.x * 256 + threadIdx.x;
  const int row = (int)((i4 * 4) / V_);
  float4* p = (float4*)y + i4;
  float4 v = *p;
  const float L = lse[row];
  v.x -= L; v.y -= L; v.z -= L; v.w -= L;
  *p = v;
}

extern "C" void kernel_launch(void* const* d_in, const int* in_sizes, int n_in,
                              void* d_out, int out_size, void* d_ws, size_t ws_size,
                              hipStream_t stream) {
  const float* target = (const float*)d_in[0];
  const float* inp    = (const float*)d_in[1];
  const float* posf   = (const float*)d_in[2];
  const float* encs   = (const float*)d_in[3];
  const float* enc    = (const float*)d_in[4];
  const float* W_ih   = (const float*)d_in[5];
  const float* W_hh   = (const float*)d_in[6];
  const float* b_ih   = (const float*)d_in[7];
  const float* b_hh   = (const float*)d_in[8];
  const float* Win_w  = (const float*)d_in[9];
  const float* Win_b  = (const float*)d_in[10];
  const float* Wout_w = (const float*)d_in[11];
  const float* Wout_b = (const float*)d_in[12];

  float* out_hs = (float*)d_out;               // [B,H]
  float* y      = (float*)d_out + B_ * H_;     // [T*B, V]

  float* ws   = (float*)d_ws;
  float* feat = ws;                  // 2048*768
  float* gi   = feat + MR * NG;      // 64*768
  float* gh   = gi + B_ * NG;        // 64*768
  float* hs   = gh + B_ * NG;        // 64*256
  float* lse  = hs + B_ * H_;        // 2048

  hipMemsetAsync(hs, 0, B_ * H_ * sizeof(float), stream);

  for (int t = 0; t < T_; t++) {
    step_gemm<<<12, 256, 64 * KGI * sizeof(float), stream>>>(
        target, inp, posf, encs, W_ih, W_hh, hs, gi, gh, t);
    step_att<<<B_, 256, 0, stream>>>(enc, Win_w, Win_b, b_ih, b_hh, gi, gh, hs,
                                     feat, out_hs, t);
  }

  out_gemm<<<dim3(V_ / 128, MR / 64), 256, 64 * NG * sizeof(float), stream>>>(
      feat, Wout_w, Wout_b, y);
  row_lse<<<MR, 256, 0, stream>>>(y, lse);
  sub_lse<<<(MR * (size_t)V_ / 4) / 256, 256, 0, stream>>>(y, lse);
}